// PQ_5188320494024
// MI455X (gfx1250) — compile-verified
//
#include <hip/hip_runtime.h>

typedef __attribute__((ext_vector_type(2))) float v2f;
typedef __attribute__((ext_vector_type(8))) float v8f;

#define S_SUB 8
#define K_CODE 256
#define D_CW 64
#define WAVES 8
#define ROWS_PER_WAVE 16
#define ROWS_PER_BLOCK (WAVES * ROWS_PER_WAVE)   // 128
#define N_TOTAL 32768
#define ROW_BLOCKS (N_TOTAL / ROWS_PER_BLOCK)    // 256
#define STRIDE (S_SUB * D_CW)                    // 512
#define EPSQ 1e-5f

__global__ __launch_bounds__(256) void pq_vq_kernel(
    const float* __restrict__ z,       // [32768, 512]
    const float* __restrict__ weight,  // [8, 256, 64]
    float* __restrict__ out,           // [32768, 512]
    float* __restrict__ loss_out)      // [1]
{
    __shared__ float lds_w[K_CODE * D_CW];        // 64 KB: weight[s], row-major [k][d]
    __shared__ float lds_c2[K_CODE];              // ||c_k||^2
    __shared__ float lds_stage[WAVES][256];       // per-wave 16x16 attn-numerator stage
    __shared__ float lds_x2[WAVES][16];           // per-row ||z||^2
    __shared__ float lds_inv[WAVES][16];          // per-row sum of 1/dist
    __shared__ int   lds_idx[WAVES][16];          // per-row argmin

    const int tid  = threadIdx.x;
    const int wave = tid >> 5;
    const int lane = tid & 31;
    const int l16  = lane & 15;
    const bool hi  = lane >= 16;
    const int hoff = hi ? 2 : 0;

    const int s  = blockIdx.x / ROW_BLOCKS;
    const int rb = blockIdx.x % ROW_BLOCKS;
    const int n0 = rb * ROWS_PER_BLOCK + wave * ROWS_PER_WAVE;

    // ---- stage weight[s] into LDS (coalesced float4) ----
    {
        const float4* wsrc = (const float4*)(weight + s * K_CODE * D_CW);
        float4* wdst = (float4*)lds_w;
        #pragma unroll
        for (int i = 0; i < 16; ++i)
            wdst[i * 256 + tid] = wsrc[i * 256 + tid];
    }
    __syncthreads();

    // ---- c2[k] = ||w_k||^2 (one k per thread) ----
    {
        float acc = 0.f;
        const float* wr = &lds_w[tid * D_CW];
        #pragma unroll
        for (int d = 0; d < D_CW; ++d) { float v = wr[d]; acc += v * v; }
        lds_c2[tid] = acc;
    }

    // ---- load z rows as WMMA f32 16x16x4 A-fragments (register resident) ----
    // lane l<16: row l, dims {4kc,4kc+1}; lane>=16: row l-16, dims {4kc+2,4kc+3}
    v2f a[16];
    float x2p = 0.f;
    {
        const float* zrow = z + (n0 + l16) * STRIDE + s * D_CW + hoff;
        #pragma unroll
        for (int kc = 0; kc < 16; ++kc) {
            v2f t = *(const v2f*)(zrow + 4 * kc);
            a[kc] = t;
            x2p += t.x * t.x + t.y * t.y;
        }
    }
    {
        float x2full = x2p + __shfl_xor(x2p, 16, 32);
        if (!hi) lds_x2[wave][l16] = x2full;
    }
    __syncthreads();   // covers lds_c2 + lds_x2

    v8f zw[4];
    #pragma unroll
    for (int dt = 0; dt < 4; ++dt) zw[dt] = (v8f){};

    float minv[8]; int mini[8]; float invs[8];
    #pragma unroll
    for (int r = 0; r < 8; ++r) { minv[r] = 3.4e38f; mini[r] = 0; invs[r] = 0.f; }

    float* stage = lds_stage[wave];

    // ---- main loop over 16 codeword tiles ----
    for (int kt = 0; kt < 16; ++kt) {
        // GEMM 1: xc tile = z(16x64) * w_tile(64x16) via 16 fp32 WMMAs
        v8f acc = (v8f){};
        const int col = kt * 16 + l16;
        const float* wcol = &lds_w[col * D_CW + hoff]; // B frag: same striping as A
        #pragma unroll
        for (int kc = 0; kc < 16; ++kc) {
            v2f b = *(const v2f*)(wcol + 4 * kc);
            acc = __builtin_amdgcn_wmma_f32_16x16x4_f32(
                false, a[kc], false, b, (short)0, acc, false, false);
        }

        // distances, argmin, attn numerators (C layout: vgpr r -> row r / r+8)
        const float c2v = lds_c2[col];
        float num[8];
        #pragma unroll
        for (int r = 0; r < 8; ++r) {
            const int rrow = r + (hi ? 8 : 0);
            float d2 = lds_x2[wave][rrow] + c2v - 2.0f * acc[r];
            if (d2 < minv[r]) { minv[r] = d2; mini[r] = col; }
            float dist = sqrtf(fmaxf(d2, 0.f));
            float nm = 1.0f / fmaxf(dist, EPSQ);   // softmax(-log d) == (1/d)/sum(1/d)
            invs[r] += nm;
            num[r] = nm;
        }

        // stage numerator tile C-layout -> row-major for A-fragment reuse
        #pragma unroll
        for (int r = 0; r < 8; ++r)
            stage[(r + (hi ? 8 : 0)) * 16 + l16] = num[r];
        __syncthreads();

        // GEMM 2 (fused): zw(16x64) += num(16x16) * w_tile(16x64), 16 WMMAs
        #pragma unroll
        for (int kc2 = 0; kc2 < 4; ++kc2) {
            v2f af;
            af.x = stage[l16 * 16 + 4 * kc2 + hoff];
            af.y = stage[l16 * 16 + 4 * kc2 + hoff + 1];
            const int kbase = kt * 16 + 4 * kc2 + hoff;
            #pragma unroll
            for (int dt = 0; dt < 4; ++dt) {
                v2f bf;
                bf.x = lds_w[(kbase + 0) * D_CW + dt * 16 + l16];
                bf.y = lds_w[(kbase + 1) * D_CW + dt * 16 + l16];
                zw[dt] = __builtin_amdgcn_wmma_f32_16x16x4_f32(
                    false, af, false, bf, (short)0, zw[dt], false, false);
            }
        }
        __syncthreads();
    }

    // ---- cross-lane reductions: masks 1..8 reduce both 16-lane halves ----
    #pragma unroll
    for (int r = 0; r < 8; ++r) {
        float v = minv[r]; int i = mini[r]; float sm = invs[r];
        #pragma unroll
        for (int m = 1; m <= 8; m <<= 1) {
            float vo = __shfl_xor(v, m, 32);
            int   io = __shfl_xor(i, m, 32);
            float so = __shfl_xor(sm, m, 32);
            sm += so;
            if (vo < v || (vo == v && io < i)) { v = vo; i = io; }
        }
        mini[r] = i; invs[r] = sm;
    }
    if (l16 == 0) {
        #pragma unroll
        for (int r = 0; r < 8; ++r) {
            lds_idx[wave][r + (hi ? 8 : 0)] = mini[r];
            lds_inv[wave][r + (hi ? 8 : 0)] = invs[r];
        }
    }
    __syncthreads();

    // ---- loss_w: normalize zw, compare to gathered codeword ----
    float lsum = 0.f;
    #pragma unroll
    for (int dt = 0; dt < 4; ++dt) {
        #pragma unroll
        for (int r = 0; r < 8; ++r) {
            const int rrow = r + (hi ? 8 : 0);
            float zwv = zw[dt][r] / lds_inv[wave][rrow];
            float zqv = lds_w[lds_idx[wave][rrow] * D_CW + dt * 16 + l16];
            float df = zwv - zqv;
            lsum += df * df;
        }
    }

    // ---- loss_q: from register-resident z fragments ----
    {
        const int idx = lds_idx[wave][l16];
        const float* zq = &lds_w[idx * D_CW + hoff];
        #pragma unroll
        for (int kc = 0; kc < 16; ++kc) {
            float d0 = a[kc].x - zq[4 * kc];
            float d1 = a[kc].y - zq[4 * kc + 1];
            lsum += d0 * d0 + d1 * d1;
        }
    }

    // ---- write z_q (straight-through forward value) ----
    #pragma unroll
    for (int r = 0; r < 16; ++r) {
        const int idx = lds_idx[wave][r];
        v2f val = *(const v2f*)&lds_w[idx * D_CW + 2 * lane];
        *(v2f*)(out + (n0 + r) * STRIDE + s * D_CW + 2 * lane) = val;
    }

    // ---- loss reduction: wave shuffle tree + one atomic per wave ----
    #pragma unroll
    for (int m = 1; m < 32; m <<= 1) lsum += __shfl_xor(lsum, m, 32);
    if (lane == 0)
        atomicAdd(loss_out, lsum * (1.0f / 16777216.0f)); // /= S*N*D
}

extern "C" void kernel_launch(void* const* d_in, const int* in_sizes, int n_in,
                              void* d_out, int out_size, void* d_ws, size_t ws_size,
                              hipStream_t stream) {
    (void)in_sizes; (void)n_in; (void)d_ws; (void)ws_size;
    const float* z      = (const float*)d_in[0];
    const float* weight = (const float*)d_in[1];
    float* out  = (float*)d_out;
    float* loss = (float*)d_out + (out_size - 1);  // last element is the scalar loss

    hipMemsetAsync(loss, 0, sizeof(float), stream);

    dim3 grid(S_SUB * ROW_BLOCKS);   // 8 * 256 = 2048 workgroups
    dim3 block(256);                 // 8 waves of 32
    pq_vq_kernel<<<grid, block, 0, stream>>>(z, weight, out, loss);
}